// Attention_71193377898866
// MI455X (gfx1250) — compile-verified
//
#include <hip/hip_runtime.h>
#include <hip/hip_bf16.h>

typedef __attribute__((ext_vector_type(2))) float v2f;
typedef __attribute__((ext_vector_type(8))) float v8f;

#define ENC_DIM 512
#define ATT_DIM 512
#define NBATCH  64
#define SEQLEN  1024
#define NROWS   (NBATCH * SEQLEN)      // 65536
#define ROWS_PER_WG 64
#define LDS_STRIDE  516                // 512 + pad: avoids 64-bank conflicts

// CDNA5 async direct-to-LDS copy (GLOBAL_LOAD_ASYNC_TO_LDS_B128, ASYNCcnt).
// vdst = 32-bit LDS byte address, vaddr = 64-bit global address ("GV" mode).
__device__ __forceinline__ void async_copy_b128(const float* gsrc, float* lds_dst) {
    const unsigned           lds32 = (unsigned)(size_t)lds_dst;   // low 32b = LDS offset
    const unsigned long long g64   = (unsigned long long)(size_t)gsrc;
    asm volatile("global_load_async_to_lds_b128 %0, %1, off"
                 :
                 : "v"(lds32), "v"(g64)
                 : "memory");
}

__device__ __forceinline__ void wait_async0() {
#if __has_builtin(__builtin_amdgcn_s_wait_asynccnt)
    __builtin_amdgcn_s_wait_asynccnt(0);
#else
    asm volatile("s_wait_asynccnt 0x0" ::: "memory");
#endif
}

// ---------------------------------------------------------------------------
// Phase 0: biasvec[b][a] = b_enc[a] + b_dec[a] + b_cov[a] + sum_d h[b,d]*W_dec[d,a]
// ---------------------------------------------------------------------------
__global__ __launch_bounds__(256)
void prep_kernel(const float* __restrict__ h,      // [64][512]
                 const float* __restrict__ W_dec,  // [512][512]
                 const float* __restrict__ b_enc,
                 const float* __restrict__ b_dec,
                 const float* __restrict__ b_cov,
                 float* __restrict__ biasvec)      // [64][512]
{
    const int b   = blockIdx.x;
    const int tid = threadIdx.x;
    for (int a = tid; a < ATT_DIM; a += 256) {
        float acc = b_enc[a] + b_dec[a] + b_cov[a];
        const float* hb = h + b * ENC_DIM;
        for (int d = 0; d < ENC_DIM; ++d)
            acc = fmaf(hb[d], W_dec[(size_t)d * ATT_DIM + a], acc);
        biasvec[b * ATT_DIM + a] = acc;
    }
}

// ---------------------------------------------------------------------------
// Phase 1: energy[row] via fused GEMM (enc @ W_enc) + bias + ReLU + dot(W_full)
// One workgroup = 64 rows, 8 waves x 64 N-cols each, fp32 WMMA 16x16x4.
// A strip staged to LDS via CDNA5 async direct-to-LDS loads (ASYNCcnt).
// ---------------------------------------------------------------------------
__global__ __launch_bounds__(256)
void energy_kernel(const float* __restrict__ enc,      // [65536][512]
                   const float* __restrict__ W_enc,    // [512][512]
                   const float* __restrict__ W_cov,    // [512]
                   const float* __restrict__ W_full,   // [512]
                   const float* __restrict__ coverage, // [65536]
                   const float* __restrict__ biasvec,  // [64][512]
                   const float* __restrict__ b_full,   // [1]
                   float* __restrict__ energy)         // [65536]
{
    __shared__ __align__(16) float A_s[ROWS_PER_WG * LDS_STRIDE]; // 132 KB
    __shared__ float cov_s[ROWS_PER_WG];
    __shared__ float e_s[ROWS_PER_WG];

    const int wg   = blockIdx.x;             // 0..1023
    const int row0 = wg * ROWS_PER_WG;
    const int b    = row0 >> 10;             // 1024 rows per batch; 64|1024
    const int tid  = threadIdx.x;
    const int wave = tid >> 5;               // 0..7
    const int lane = tid & 31;
    const int half = lane >> 4;              // lane half selects K pair
    const int l16  = lane & 15;

    // Stage the 64x512 A strip into LDS via async direct-to-LDS b128 copies.
    const float* Abase = enc + (size_t)row0 * ENC_DIM;
    for (int i = tid; i < ROWS_PER_WG * ENC_DIM / 4; i += 256) {
        const int r  = i >> 7;               // 128 float4 per row
        const int c4 = i & 127;
        async_copy_b128(Abase + (size_t)r * ENC_DIM + c4 * 4,
                        &A_s[r * LDS_STRIDE + c4 * 4]);
    }
    wait_async0();
    if (tid < ROWS_PER_WG) {
        cov_s[tid] = coverage[row0 + tid];
        e_s[tid]   = 0.0f;
    }
    __syncthreads();

    const int n0 = wave * 64;                // this wave's 64-column N slice

    v8f acc[4][4];
    #pragma unroll
    for (int mt = 0; mt < 4; ++mt)
        #pragma unroll
        for (int nt = 0; nt < 4; ++nt)
            acc[mt][nt] = v8f{0.f,0.f,0.f,0.f,0.f,0.f,0.f,0.f};

    for (int k0 = 0; k0 < ENC_DIM; k0 += 4) {
        const int kb = k0 + half * 2;        // ISA A/B layout: lane half -> K pair
        // B fragments (W_enc is L2-resident: 1 MB shared by all workgroups)
        v2f bf[4];
        #pragma unroll
        for (int nt = 0; nt < 4; ++nt) {
            const int n = n0 + nt * 16 + l16;
            bf[nt].x = W_enc[(size_t)kb       * ATT_DIM + n];
            bf[nt].y = W_enc[(size_t)(kb + 1) * ATT_DIM + n];
        }
        #pragma unroll
        for (int mt = 0; mt < 4; ++mt) {
            const float* ap = &A_s[(mt * 16 + l16) * LDS_STRIDE + kb];
            v2f af; af.x = ap[0]; af.y = ap[1];
            #pragma unroll
            for (int nt = 0; nt < 4; ++nt) {
                acc[mt][nt] = __builtin_amdgcn_wmma_f32_16x16x4_f32(
                    false, af, false, bf[nt], (short)0, acc[mt][nt],
                    false, false);
            }
        }
    }

    // Epilogue: bias + cov*W_cov, ReLU, dot with W_full, reduce over N
    #pragma unroll
    for (int mt = 0; mt < 4; ++mt) {
        float rowsum[8];
        #pragma unroll
        for (int r = 0; r < 8; ++r) rowsum[r] = 0.0f;
        #pragma unroll
        for (int nt = 0; nt < 4; ++nt) {
            const int n   = n0 + nt * 16 + l16;
            const float wf = W_full[n];
            const float wc = W_cov[n];
            const float bb = biasvec[b * ATT_DIM + n];
            #pragma unroll
            for (int r = 0; r < 8; ++r) {
                const int m = mt * 16 + r + 8 * half;   // C/D layout row
                float v = acc[mt][nt][r] + bb + cov_s[m] * wc;
                v = fmaxf(v, 0.0f);
                rowsum[r] = fmaf(v, wf, rowsum[r]);
            }
        }
        #pragma unroll
        for (int r = 0; r < 8; ++r) {
            const int m = mt * 16 + r + 8 * half;
            atomicAdd(&e_s[m], rowsum[r]);              // ds_add_f32
        }
    }
    __syncthreads();
    if (tid < ROWS_PER_WG)
        energy[row0 + tid] = e_s[tid] + b_full[0];
}

// ---------------------------------------------------------------------------
// Phase 2: per-batch softmax over L=1024; also emits new_coverage and zeroes awe
// ---------------------------------------------------------------------------
__global__ __launch_bounds__(256)
void softmax_kernel(const float* __restrict__ energy,
                    const float* __restrict__ coverage,
                    float* __restrict__ alpha,
                    float* __restrict__ newcov,
                    float* __restrict__ awe)
{
    __shared__ float red[8];
    const int b    = blockIdx.x;
    const int tid  = threadIdx.x;
    const int lane = tid & 31;
    const int wave = tid >> 5;

    float v[4];
    float vmax = -3.4e38f;
    #pragma unroll
    for (int i = 0; i < 4; ++i) {
        v[i] = energy[b * SEQLEN + i * 256 + tid];
        vmax = fmaxf(vmax, v[i]);
    }
    #pragma unroll
    for (int off = 16; off > 0; off >>= 1)
        vmax = fmaxf(vmax, __shfl_xor(vmax, off, 32));
    if (lane == 0) red[wave] = vmax;
    __syncthreads();
    vmax = red[0];
    #pragma unroll
    for (int w = 1; w < 8; ++w) vmax = fmaxf(vmax, red[w]);

    float s = 0.0f;
    #pragma unroll
    for (int i = 0; i < 4; ++i) { v[i] = expf(v[i] - vmax); s += v[i]; }
    #pragma unroll
    for (int off = 16; off > 0; off >>= 1)
        s += __shfl_xor(s, off, 32);
    __syncthreads();
    if (lane == 0) red[wave] = s;
    __syncthreads();
    s = 0.0f;
    #pragma unroll
    for (int w = 0; w < 8; ++w) s += red[w];
    const float inv = 1.0f / s;

    #pragma unroll
    for (int i = 0; i < 4; ++i) {
        const int idx = b * SEQLEN + i * 256 + tid;
        const float a = v[i] * inv;
        alpha[idx]  = a;
        newcov[idx] = coverage[idx] + a;
    }
    // zero this batch's slice of the awe output (d_out is poisoned 0xAA)
    awe[b * ENC_DIM + tid]       = 0.0f;
    awe[b * ENC_DIM + 256 + tid] = 0.0f;
}

// ---------------------------------------------------------------------------
// Phase 3: awe[b,e] = sum_l enc[b,l,e] * alpha[b,l]  (L split over 8 chunks)
// ---------------------------------------------------------------------------
__global__ __launch_bounds__(256)
void awe_kernel(const float* __restrict__ enc,
                const float* __restrict__ alpha,
                float* __restrict__ awe)
{
    __shared__ __align__(16) float a_s[128];
    const int b   = blockIdx.x;
    const int lc  = blockIdx.y;      // 0..7, 128 l's each
    const int tid = threadIdx.x;

    if (tid < 32)
        async_copy_b128(alpha + b * SEQLEN + lc * 128 + tid * 4, &a_s[tid * 4]);
    wait_async0();
    __syncthreads();

    float acc0 = 0.0f, acc1 = 0.0f;
    const float* base = enc + ((size_t)b * SEQLEN + lc * 128) * ENC_DIM;
    for (int l = 0; l < 128; ++l) {
        const float a = a_s[l];
        acc0 = fmaf(base[(size_t)l * ENC_DIM + tid],        a, acc0);
        acc1 = fmaf(base[(size_t)l * ENC_DIM + 256 + tid],  a, acc1);
    }
    atomicAdd(&awe[b * ENC_DIM + tid],       acc0);
    atomicAdd(&awe[b * ENC_DIM + 256 + tid], acc1);
}

// ---------------------------------------------------------------------------
extern "C" void kernel_launch(void* const* d_in, const int* in_sizes, int n_in,
                              void* d_out, int out_size, void* d_ws, size_t ws_size,
                              hipStream_t stream) {
    const float* enc      = (const float*)d_in[0];   // [64,1024,512]
    const float* h        = (const float*)d_in[1];   // [64,512]
    const float* coverage = (const float*)d_in[2];   // [64,1024]
    const float* W_enc    = (const float*)d_in[3];   // [512,512]
    const float* b_enc    = (const float*)d_in[4];   // [512]
    const float* W_dec    = (const float*)d_in[5];   // [512,512]
    const float* b_dec    = (const float*)d_in[6];   // [512]
    const float* W_cov    = (const float*)d_in[7];   // [1,512] -> flat 512
    const float* b_cov    = (const float*)d_in[8];   // [512]
    const float* W_full   = (const float*)d_in[9];   // [512,1] -> flat 512
    const float* b_full   = (const float*)d_in[10];  // [1]

    float* biasvec = (float*)d_ws;                                  // 64*512
    float* energy  = (float*)((char*)d_ws + NBATCH * ATT_DIM * 4);  // 65536

    float* out    = (float*)d_out;
    float* awe    = out;                              // [64,512]
    float* alpha  = out + NBATCH * ENC_DIM;           // [64,1024]
    float* newcov = alpha + NBATCH * SEQLEN;          // [64,1024]

    prep_kernel<<<NBATCH, 256, 0, stream>>>(h, W_dec, b_enc, b_dec, b_cov, biasvec);
    energy_kernel<<<NROWS / ROWS_PER_WG, 256, 0, stream>>>(
        enc, W_enc, W_cov, W_full, coverage, biasvec, b_full, energy);
    softmax_kernel<<<NBATCH, 256, 0, stream>>>(energy, coverage, alpha, newcov, awe);
    awe_kernel<<<dim3(NBATCH, 8), 256, 0, stream>>>(enc, alpha, awe);
}